// ContrastiveLoss_23287312679410
// MI455X (gfx1250) — compile-verified
//
#include <hip/hip_runtime.h>
#include <hip/hip_bf16.h>
#include <math.h>

typedef __attribute__((ext_vector_type(2))) float v2f;
typedef __attribute__((ext_vector_type(8))) float v8f;

#define B_     8
#define C_     128
#define H_     32
#define W_     32
#define HW_    1024
#define NEG_   256
#define TEMP_  2.0f
#define FACTOR_ 0.8f
#define EPS_   1e-8f
#define ROWS_  16

// workspace layout (float units):
//   [0      .. B_*257)        acc[b][0..256]   (zeroed by prep)
//   [4096   .. 4096+8192)     n1[b][h]
//   [12288  .. 12288+8192)    nn[b][p]
#define WS_ACC 0
#define WS_N1  4096
#define WS_NN  12288

// ---------------------------------------------------------------- prep ----
__global__ __launch_bounds__(256) void cl_prep_kernel(
    const float* __restrict__ v1, const float* __restrict__ v2,
    float* __restrict__ ws) {
  int t = blockIdx.x * blockDim.x + threadIdx.x;
  if (t < B_ * 257) ws[WS_ACC + t] = 0.0f;
  if (t < B_ * HW_) {
    const float* p = v1 + (t / HW_) * C_ * HW_ + (t % HW_);
    float s = 0.0f;
    for (int c = 0; c < C_; ++c) { float x = p[c * HW_]; s += x * x; }
    ws[WS_N1 + t] = sqrtf(s);
  } else if (t < 2 * B_ * HW_) {
    int u = t - B_ * HW_;
    const float* p = v2 + (u / HW_) * C_ * HW_ + (u % HW_);
    float s = 0.0f;
    for (int c = 0; c < C_; ++c) { float x = p[c * HW_]; s += x * x; }
    ws[WS_NN + u] = sqrtf(s);
  }
}

// ---------------------------------------------------------------- main ----
// Grid: B_ * (HW_/ROWS_) = 8*64 = 512 blocks of 256 threads (8 wave32).
// Dynamic LDS: gram[ROWS_][HW_] = 65536 bytes.
extern __shared__ float cl_smem[];

__global__ __launch_bounds__(256) void cl_main_kernel(
    const float* __restrict__ v1, const float* __restrict__ v2,
    const float* __restrict__ img, const int* __restrict__ nidx,
    float* __restrict__ ws) {
  const int b    = blockIdx.x >> 6;   // /64 row tiles
  const int tile = blockIdx.x & 63;
  const int h0   = tile * ROWS_;

  float* gram = cl_smem;              // ROWS_ x HW_

  const float* z1 = v1 + b * (C_ * HW_);
  const float* z2 = v2 + b * (C_ * HW_);

  // ---------------- phase 1: Gram strip via V_WMMA_F32_16X16X4_F32 -------
  const int lane = threadIdx.x & 31;
  const int wave = threadIdx.x >> 5;
  const int m    = lane & 15;           // row (A) / col (B) within tile
  const int kb   = (lane >> 4) << 1;    // K base: lanes 0-15 -> {0,1}, 16-31 -> {2,3}

  // Preload A fragments for all 32 K-steps (16x4 f32 layout, ISA 7.12.2)
  v2f afrag[32];
#pragma unroll
  for (int kk = 0; kk < 32; ++kk) {
    int k = kk * 4 + kb;
    afrag[kk].x = z1[k * HW_ + h0 + m];
    afrag[kk].y = z1[(k + 1) * HW_ + h0 + m];
  }

  for (int t = 0; t < 8; ++t) {         // 8 column tiles per wave
    int p0 = (wave * 8 + t) * 16;
    v8f c = {};
#pragma unroll
    for (int kk = 0; kk < 32; ++kk) {   // K = 128 in steps of 4
      int k = kk * 4 + kb;
      v2f bfrag;
      bfrag.x = z2[k * HW_ + p0 + m];
      bfrag.y = z2[(k + 1) * HW_ + p0 + m];
      c = __builtin_amdgcn_wmma_f32_16x16x4_f32(
          false, afrag[kk], false, bfrag, (short)0, c, false, false);
    }
    // scatter D to LDS: VGPR r holds M=r (lanes 0-15) / M=r+8 (lanes 16-31)
    int half = lane >> 4;
#pragma unroll
    for (int r = 0; r < 8; ++r) {
      int mm = r + 8 * half;
      gram[mm * HW_ + p0 + m] = c[r];
    }
  }
  __syncthreads();

  // ---------------- phase 2: gather + weight + accumulate -----------------
  const float inv_diag = 1.0f / sqrtf((float)((H_ - 1) * (H_ - 1) + (W_ - 1) * (W_ - 1)));
  const float inv_s3   = 1.0f / sqrtf(3.0f);
  const int   n        = threadIdx.x;   // 0..255 (one neg column per thread)
  float*      accb     = ws + WS_ACC + b * 257;
  const float* n1b     = ws + WS_N1 + b * HW_;
  const float* nnb     = ws + WS_NN + b * HW_;

  float simSum = 0.0f;
  for (int hl = 0; hl < ROWS_; ++hl) {
    int   h   = h0 + hl;
    float hhf = (float)(h >> 5);
    float wwf = (float)(h & 31);
    float r0  = img[0 * HW_ + h];
    float g0  = img[1 * HW_ + h];
    float bl0 = img[2 * HW_ + h];

    int nh = nidx[((b * 2 + 0) * HW_ + h) * NEG_ + n];
    int nw = nidx[((b * 2 + 1) * HW_ + h) * NEG_ + n];
    int p  = nh * W_ + nw;

    float dh  = hhf - (float)nh;
    float dw  = wwf - (float)nw;
    float euc = sqrtf(dh * dh + dw * dw) * inv_diag;

    float dr  = r0 - img[0 * HW_ + p];
    float dg  = g0 - img[1 * HW_ + p];
    float db  = bl0 - img[2 * HW_ + p];
    float rgb = sqrtf(dr * dr + dg * dg + db * db) * inv_s3;

    float weight = euc * FACTOR_ + rgb * (1.0f - FACTOR_);

    float dot   = gram[hl * HW_ + p];
    float denom = fmaxf(n1b[h], EPS_) * fmaxf(nnb[p], EPS_);
    float sim   = fabsf(dot / denom * weight);
    simSum += fminf(sim, 1.0f);
  }
  atomicAdd(&accb[1 + n], simSum);

  if (threadIdx.x < ROWS_) {            // sim0 term, one per row
    float v  = n1b[h0 + threadIdx.x];
    float mx = fmaxf(v, EPS_);
    float s0 = fminf(fabsf((v * v) / (mx * mx)), 1.0f);
    atomicAdd(&accb[0], s0);
  }
}

// --------------------------------------------------------------- final ----
__global__ __launch_bounds__(256) void cl_final_kernel(
    const float* __restrict__ ws, float* __restrict__ out) {
  __shared__ float sLog[256];
  __shared__ float sS[256];
  int j = threadIdx.x;
  float slog = 0.0f, ss = 0.0f;
  for (int b = 0; b < B_; ++b) {
    float s = ws[WS_ACC + b * 257 + 1 + j] * (1.0f / (float)HW_) * (1.0f / TEMP_);
    ss += s;
    slog += fmaxf(logf(1.0f - s), -100.0f);
  }
  sLog[j] = slog;
  sS[j]   = ss;
  __syncthreads();
  if (j == 0) {
    float totLog = 0.0f, totS = 0.0f;
    for (int k = 0; k < 256; ++k) { totLog += sLog[k]; totS += sS[k]; }
    float sumLogP0 = 0.0f, sumSim0 = 0.0f;
    for (int b = 0; b < B_; ++b) {
      float s0 = ws[WS_ACC + b * 257] * (1.0f / (float)HW_);
      sumSim0  += s0;
      sumLogP0 += fmaxf(logf(s0), -100.0f);
    }
    out[0] = (-sumLogP0 - totLog) / (257.0f * (float)B_);
    out[1] = sumSim0 / (float)B_;
    out[2] = totS / (float)NEG_ * TEMP_ / (float)B_;
  }
}

// -------------------------------------------------------------- launch ----
extern "C" void kernel_launch(void* const* d_in, const int* in_sizes, int n_in,
                              void* d_out, int out_size, void* d_ws, size_t ws_size,
                              hipStream_t stream) {
  const float* v1   = (const float*)d_in[0];
  const float* v2   = (const float*)d_in[1];
  const float* img  = (const float*)d_in[2];
  const int*   nidx = (const int*)d_in[3];
  float*       ws   = (float*)d_ws;
  float*       out  = (float*)d_out;

  // 1) zero accumulators + column norms
  cl_prep_kernel<<<(2 * B_ * HW_) / 256, 256, 0, stream>>>(v1, v2, ws);

  // 2) WMMA Gram strips + weighted gather-accumulate
  size_t smem = (size_t)ROWS_ * HW_ * sizeof(float);   // 65536 B
  cl_main_kernel<<<B_ * (HW_ / ROWS_), 256, smem, stream>>>(v1, v2, img, nidx, ws);

  // 3) fold to the 3 scalar outputs
  cl_final_kernel<<<1, 256, 0, stream>>>(ws, out);
}